// _AceStepResidualFSQ_23776938950706
// MI455X (gfx1250) — compile-verified
//
#include <hip/hip_runtime.h>

typedef __attribute__((ext_vector_type(2))) float v2f;
typedef __attribute__((ext_vector_type(8))) float v8f;

#define D_DIM 2048
#define C_DIM 6

// ---------------------------------------------------------------------------
// Kernel 0: zero-pad w_in [6][2048] -> winp [16][2048] so WMMA B-columns 6..15
// read exact zeros with unconditional loads.
// ---------------------------------------------------------------------------
__global__ __launch_bounds__(256) void fsq_prep_kernel(const float* __restrict__ w_in,
                                                       float* __restrict__ winp) {
    int i = blockIdx.x * 256 + threadIdx.x;
    if (i < 16 * D_DIM) {
        winp[i] = (i < C_DIM * D_DIM) ? w_in[i] : 0.0f;
    }
}

// ---------------------------------------------------------------------------
// Kernel 1: in-projection [tokens,2048]x[2048,6] via V_WMMA_F32_16X16X4_F32,
// tanh soft-clamp, FSQ quantize, write codes (padded to 8 ch) + indices.
// One wave = 16 tokens. Block = 256 threads = 8 waves = 128 tokens.
// ---------------------------------------------------------------------------
__global__ __launch_bounds__(256) void fsq_in_kernel(const float* __restrict__ x,
                                                     const float* __restrict__ winp,
                                                     const float* __restrict__ b_in,
                                                     float* __restrict__ codes8,
                                                     int* __restrict__ idx_out) {
    const int tid  = threadIdx.x;
    const int lane = tid & 31;
    const int wave = tid >> 5;
    const int n    = lane & 15;   // A row (token) for A-frag, channel for B/C
    const int h    = lane >> 4;   // K half selector
    const int tb   = (blockIdx.x * 8 + wave) * 16;

    // A: lane (m + 16h) holds x[token m][k + 2h], x[token m][k + 2h + 1]
    const float* xa = x + (size_t)(tb + n) * D_DIM + 2 * h;
    // B: lane (n + 16h) holds w_inT[k+2h][n], w_inT[k+2h+1][n] = winp[n][k+2h..]
    const float* wa = winp + n * D_DIM + 2 * h;

    v8f acc0 = {0.f, 0.f, 0.f, 0.f, 0.f, 0.f, 0.f, 0.f};
    v8f acc1 = {0.f, 0.f, 0.f, 0.f, 0.f, 0.f, 0.f, 0.f};

#pragma unroll 4
    for (int k = 0; k < D_DIM; k += 8) {
        v2f A0 = *(const v2f*)(xa + k);
        v2f B0 = *(const v2f*)(wa + k);
        v2f A1 = *(const v2f*)(xa + k + 4);
        v2f B1 = *(const v2f*)(wa + k + 4);
        acc0 = __builtin_amdgcn_wmma_f32_16x16x4_f32(false, A0, false, B0,
                                                     (short)0, acc0, false, false);
        acc1 = __builtin_amdgcn_wmma_f32_16x16x4_f32(false, A1, false, B1,
                                                     (short)0, acc1, false, false);
    }
    v8f acc = acc0 + acc1;  // same C layout -> elementwise add is valid

    // C layout: lane (n + 16h) element j = h[token tb+j+8h][channel n]
    const float Ln     = (n < 3) ? 8.0f : 5.0f;   // LEVELS = [8,8,8,5,5,5]
    const float lm1    = Ln - 1.0f;
    const float stepq  = 2.0f / lm1;
    const float sc     = 1.0f + 1.0f / lm1;       // soft clamp scale
    const float inv_sc = 1.0f / sc;
    const float binv   = (n < C_DIM) ? b_in[n] : 0.0f;
    const float basisf = (n == 0) ? 1.0f
                       : (n == 1) ? 8.0f
                       : (n == 2) ? 64.0f
                       : (n == 3) ? 512.0f
                       : (n == 4) ? 2560.0f
                       : (n == 5) ? 12800.0f : 0.0f;

#pragma unroll
    for (int j = 0; j < 8; ++j) {
        float hv  = acc[j] + binv;
        hv        = tanhf(hv * inv_sc) * sc;
        float xcl = fminf(fmaxf(hv, -1.0f), 1.0f);
        float br  = lm1 * (xcl + 1.0f) * 0.5f + 0.5f;
        float li  = floorf(br);               // level index (float, exact int)
        float q   = stepq * li - 1.0f;        // quantized code

        const int tok = tb + j + 8 * h;
        if (n < 8) codes8[tok * 8 + n] = (n < C_DIM) ? q : 0.0f;

        // FSQ index = sum_c li_c * basis_c ; reduce within each 16-lane half
        float contrib = li * basisf;
        contrib += __shfl_xor(contrib, 1);
        contrib += __shfl_xor(contrib, 2);
        contrib += __shfl_xor(contrib, 4);
        contrib += __shfl_xor(contrib, 8);
        if (n == 0) idx_out[tok] = __float2int_rn(contrib);
    }
}

// ---------------------------------------------------------------------------
// Kernel 2: out-projection [tokens,6]x[6,2048] via V_WMMA_F32_16X16X4_F32.
// One wave = 16 tokens x all 2048 d. w_outT fragments for all 128 d-slices
// pre-staged in 64KB LDS (conflict-free per-lane v2f reads). C initialized
// with b_out so bias accumulates for free.
// ---------------------------------------------------------------------------
__global__ __launch_bounds__(256) void fsq_out_kernel(const float* __restrict__ codes8,
                                                      const float* __restrict__ w_out,
                                                      const float* __restrict__ b_out,
                                                      float* __restrict__ out) {
    extern __shared__ char smem[];
    v2f* fragB0 = (v2f*)smem;             // [128 slices][32 lanes], K-chunk 0 (c=0..3)
    v2f* fragB1 = (v2f*)(smem + 32768);   // [128 slices][32 lanes], K-chunk 1 (c=4,5 + 0)

    const int tid = threadIdx.x;
    // Stage B fragments: lane (n + 16h) of slice s holds w_out[d=s*16+n][k+2h .. k+2h+1]
    for (int e = tid; e < 128 * 32; e += 256) {
        int s  = e >> 5;
        int L  = e & 31;
        int n2 = L & 15;
        int hh = L >> 4;
        int d  = s * 16 + n2;
        fragB0[e] = *(const v2f*)(w_out + d * C_DIM + 2 * hh);      // c = {0,1} / {2,3}
        v2f z; z[0] = 0.0f; z[1] = 0.0f;
        fragB1[e] = (hh == 0) ? *(const v2f*)(w_out + d * C_DIM + 4) // c = {4,5}
                              : z;                                    // c = {6,7} -> 0
    }
    __syncthreads();

    const int lane = tid & 31;
    const int wave = tid >> 5;
    const int m    = lane & 15;
    const int h    = lane >> 4;
    const int tb   = (blockIdx.x * 8 + wave) * 16;

    // A fragments (codes, K padded 6->8 with zeros already in codes8)
    const float* crow = codes8 + (size_t)(tb + m) * 8;
    v2f a0 = *(const v2f*)(crow + 2 * h);       // c {0,1}/{2,3}
    v2f a1 = *(const v2f*)(crow + 4 + 2 * h);   // c {4,5}/{6,7}(=0)

    float* obase = out + (size_t)(tb + 8 * h) * D_DIM + m;

    for (int s = 0; s < 128; ++s) {
        const int dbase = s * 16;
        const float bval = b_out[dbase + m];
        v8f c = {bval, bval, bval, bval, bval, bval, bval, bval};

        v2f b0 = fragB0[s * 32 + lane];
        v2f b1 = fragB1[s * 32 + lane];
        c = __builtin_amdgcn_wmma_f32_16x16x4_f32(false, a0, false, b0,
                                                  (short)0, c, false, false);
        c = __builtin_amdgcn_wmma_f32_16x16x4_f32(false, a1, false, b1,
                                                  (short)0, c, false, false);

        float* p = obase + dbase;
#pragma unroll
        for (int j = 0; j < 8; ++j) {
            p[(size_t)j * D_DIM] = c[j];  // token tb+j+8h, d = dbase + m
        }
    }
}

// ---------------------------------------------------------------------------
extern "C" void kernel_launch(void* const* d_in, const int* in_sizes, int n_in,
                              void* d_out, int out_size, void* d_ws, size_t ws_size,
                              hipStream_t stream) {
    (void)n_in; (void)out_size; (void)ws_size;
    const float* hidden = (const float*)d_in[0];
    const float* w_in   = (const float*)d_in[1];
    const float* b_in   = (const float*)d_in[2];
    const float* w_out  = (const float*)d_in[3];
    const float* b_out  = (const float*)d_in[4];

    const int n_hidden = in_sizes[0];          // B*S*D = 67108864
    const int tokens   = n_hidden / D_DIM;     // 32768

    float* out     = (float*)d_out;
    int*   idx_out = (int*)d_out + n_hidden;   // indices follow `out` flat

    // Workspace layout: padded w_in [16*2048] then codes8 [tokens*8]
    float* winp   = (float*)d_ws;
    float* codes8 = winp + 16 * D_DIM;

    fsq_prep_kernel<<<(16 * D_DIM + 255) / 256, 256, 0, stream>>>(w_in, winp);

    const int blocks = tokens / 128;           // 128 tokens per block (8 waves)
    fsq_in_kernel<<<blocks, 256, 0, stream>>>(hidden, winp, b_in, codes8, idx_out);
    fsq_out_kernel<<<blocks, 256, 65536, stream>>>(codes8, w_out, b_out, out);
}